// VanillaGNNLayer_69913477644665
// MI455X (gfx1250) — compile-verified
//
#include <hip/hip_runtime.h>

typedef __attribute__((ext_vector_type(2))) float v2f;
typedef __attribute__((ext_vector_type(8))) float v8f;

#define D 128           // D_IN == D_OUT == 128
#define WAVES_PER_BLOCK 8

// ---------------------------------------------------------------------------
// Zero the output accumulator (harness poisons d_out with 0xAA).
// ---------------------------------------------------------------------------
__global__ __launch_bounds__(256) void zero_f4(float4* __restrict__ p, int n4) {
    int i = blockIdx.x * blockDim.x + threadIdx.x;
    if (i < n4) p[i] = make_float4(0.f, 0.f, 0.f, 0.f);
}

// ---------------------------------------------------------------------------
// h = x @ W^T using V_WMMA_F32_16X16X4_F32 (fp32 matrix path, wave32).
// One wave computes a 16(M) x 128(N) tile of h.
//
// A-fragment layout (32-bit A 16x4, ISA 7.12.2):
//   lanes 0-15 : VGPR0=K0, VGPR1=K1 for row M=lane
//   lanes 16-31: VGPR0=K2, VGPR1=K3 for row M=lane-16
// -> per lane: one 8B load of x[row0+lane%16][k0 + (lane>=16 ? 2 : 0)..+1]
//
// B-fragment (4x16, B[k][n] = W[n][k]):
//   lanes 0-15 : rows K0,K1 at col N=lane  -> W[n][k0+0..1]
//   lanes 16-31: rows K2,K3 at col N=lane-16 -> W[n][k0+2..3]
//
// C/D layout: VGPR r holds M=r (lanes 0-15) / M=r+8 (lanes 16-31), N=lane%16.
//
// x is a single-pass stream (51.2 MB) -> non-temporal loads so it does not
// evict h/out from the 192MB L2 ahead of the scatter phase. W (64 KB, reused
// by every wave) stays regular-temporal.
// ---------------------------------------------------------------------------
__global__ __launch_bounds__(256) void gemm_xwt_wmma(const float* __restrict__ x,
                                                     const float* __restrict__ w,
                                                     float* __restrict__ h,
                                                     int mTiles) {
    const int lane = threadIdx.x & 31;
    const int wave = threadIdx.x >> 5;
    const int tile = blockIdx.x * WAVES_PER_BLOCK + wave;   // wave-uniform
    if (tile >= mTiles) return;                             // uniform branch: EXEC stays all-1s

    const int laneLo = lane & 15;
    const int kOff   = (lane >> 4) << 1;   // 0 for lanes 0-15, 2 for lanes 16-31
    const int row0   = tile << 4;

    const float* aBase = x + (size_t)(row0 + laneLo) * D + kOff;

    v8f acc[8] = {};   // 8 N-tiles of 16 -> full 128-wide output row block

    for (int k0 = 0; k0 < D; k0 += 4) {
        v2f a = __builtin_nontemporal_load((const v2f*)(aBase + k0));
#pragma unroll
        for (int nt = 0; nt < 8; ++nt) {
            const float* bp = w + (size_t)((nt << 4) + laneLo) * D + kOff + k0;
            v2f b = *(const v2f*)bp;
            // 8 args: (neg_a, A, neg_b, B, c_mod, C, reuse_a, reuse_b)
            acc[nt] = __builtin_amdgcn_wmma_f32_16x16x4_f32(
                false, a, false, b, (short)0, acc[nt], false, false);
        }
    }

    const int mOff = (lane >> 4) << 3;     // lanes 16-31 hold rows M=8..15
#pragma unroll
    for (int nt = 0; nt < 8; ++nt) {
        float* hp = h + (size_t)(row0 + mOff) * D + (nt << 4) + laneLo;
#pragma unroll
        for (int r = 0; r < 8; ++r) {
            hp[(size_t)r * D] = acc[nt][r];   // h is reused by scatter: keep RT
        }
    }
}

// Scalar fallback for a non-multiple-of-16 row tail (not hit for N=100000).
__global__ void gemm_tail(const float* __restrict__ x, const float* __restrict__ w,
                          float* __restrict__ h, int rowStart, int nRows) {
    if ((int)blockIdx.x >= nRows) return;
    const int row = rowStart + blockIdx.x;
    const int col = threadIdx.x;   // blockDim.x == 128
    float s = 0.f;
    for (int k = 0; k < D; ++k) s += x[(size_t)row * D + k] * w[(size_t)col * D + k];
    h[(size_t)row * D + col] = s;
}

// ---------------------------------------------------------------------------
// Edge scatter: out[dst] += h[src]. One wave per edge.
// Each lane gathers a float4 (global_load_b128, 512B coalesced per wave) and
// issues 4 non-returning relaxed agent-scope f32 atomics -> resolved at L2
// (h + out, 102 MB combined, fit in the 192MB L2).
// Edge index arrays are a single-pass stream -> non-temporal loads.
// ---------------------------------------------------------------------------
__global__ __launch_bounds__(256) void scatter_add(const float* __restrict__ h,
                                                   const int* __restrict__ src,
                                                   const int* __restrict__ dst,
                                                   float* __restrict__ out,
                                                   int nEdges) {
    const int lane = threadIdx.x & 31;
    const int edge = (int)(((size_t)blockIdx.x * blockDim.x + threadIdx.x) >> 5);
    if (edge >= nEdges) return;   // wave-uniform

    const int s = __builtin_amdgcn_readfirstlane(__builtin_nontemporal_load(src + edge));
    const int d = __builtin_amdgcn_readfirstlane(__builtin_nontemporal_load(dst + edge));

    const float4 v = *(const float4*)(h + (size_t)s * D + (lane << 2));
    float* o = out + (size_t)d * D + (lane << 2);

    __hip_atomic_fetch_add(o + 0, v.x, __ATOMIC_RELAXED, __HIP_MEMORY_SCOPE_AGENT);
    __hip_atomic_fetch_add(o + 1, v.y, __ATOMIC_RELAXED, __HIP_MEMORY_SCOPE_AGENT);
    __hip_atomic_fetch_add(o + 2, v.z, __ATOMIC_RELAXED, __HIP_MEMORY_SCOPE_AGENT);
    __hip_atomic_fetch_add(o + 3, v.w, __ATOMIC_RELAXED, __HIP_MEMORY_SCOPE_AGENT);
}

extern "C" void kernel_launch(void* const* d_in, const int* in_sizes, int n_in,
                              void* d_out, int out_size, void* d_ws, size_t ws_size,
                              hipStream_t stream) {
    const float* x  = (const float*)d_in[0];   // [N_NODES, 128]
    const float* w  = (const float*)d_in[1];   // [128, 128]
    const int*   ei = (const int*)d_in[2];     // [2, N_EDGES]
    float* out = (float*)d_out;                // [N_NODES, 128]
    float* h   = (float*)d_ws;                 // scratch: [N_NODES, 128] fp32

    const int nNodes = in_sizes[0] / D;
    const int nEdges = in_sizes[2] / 2;
    const int* src = ei;
    const int* dst = ei + nEdges;

    // 1) zero the accumulator
    const int n4 = out_size / 4;
    zero_f4<<<(n4 + 255) / 256, 256, 0, stream>>>((float4*)out, n4);

    // 2) h = x @ W^T via fp32 WMMA
    const int mTiles = nNodes / 16;
    if (mTiles > 0) {
        gemm_xwt_wmma<<<(mTiles + WAVES_PER_BLOCK - 1) / WAVES_PER_BLOCK, 256, 0, stream>>>(
            x, w, h, mTiles);
    }
    const int rem = nNodes - mTiles * 16;
    if (rem > 0) {
        gemm_tail<<<rem, D, 0, stream>>>(x, w, h, mTiles * 16, rem);
    }

    // 3) out[dst] += h[src] via L2 atomics, one wave per edge
    const int blocks = (int)(((size_t)nEdges + WAVES_PER_BLOCK - 1) / WAVES_PER_BLOCK);
    scatter_add<<<blocks, 256, 0, stream>>>(h, src, dst, out, nEdges);
}